// GCN_21474836480575
// MI455X (gfx1250) — compile-verified
//
#include <hip/hip_runtime.h>

typedef float v2f __attribute__((ext_vector_type(2)));
typedef float v8f __attribute__((ext_vector_type(8)));

#define GCN_F_IN 54
#define GCN_H    16

// ---------------- degree / normalization ----------------

__global__ void k_init_deg(float* __restrict__ deg, int n) {
  int i = blockIdx.x * blockDim.x + threadIdx.x;
  if (i < n) deg[i] = 1.0f;  // self-loop contributes 1 to every node's degree
}

__global__ void k_count_deg(const long long* __restrict__ dst, int E,
                            float* __restrict__ deg) {
  int e = blockIdx.x * blockDim.x + threadIdx.x;
  if (e < E) atomicAdd(&deg[(int)dst[e]], 1.0f);
}

__global__ void k_rsqrt_inplace(float* __restrict__ deg, int n) {
  int i = blockIdx.x * blockDim.x + threadIdx.x;
  if (i < n) deg[i] = rsqrtf(deg[i]);  // deg >= 1 always
}

__global__ void k_zero(float* __restrict__ p, int n) {
  int i = blockIdx.x * blockDim.x + threadIdx.x;
  if (i < n) p[i] = 0.0f;
}

// ---------------- WMMA GEMM: D[M x 16] = A[M x K] * W[K x 16] ----------------
// K is compile-time: full steps use unconditional b64 (A) / b32 (B) loads;
// only the K%4 tail step (K=54) uses clamped-address loads + branchless
// v_cndmask zeroing. EXEC stays all-1s; no control flow inside the loop.
// Requires M % 16 == 0 (grid sized to exactly M/16 tiles).

template <int K>
__global__ void k_gemm_wmma_f32(const float* __restrict__ A,
                                const float* __restrict__ W,
                                float* __restrict__ D, int ntiles) {
  const int tile = blockIdx.x * (blockDim.x >> 5) + (threadIdx.x >> 5);
  if (tile >= ntiles) return;  // wave-uniform exit

  const int lane = threadIdx.x & 31;
  const int hi   = lane >> 4;   // half-wave select
  const int nl   = lane & 15;

  const long long row = (long long)tile * 16 + nl;
  const float* __restrict__ arow = A + row * K;

  v8f c = {};
  constexpr int KS = (K + 3) / 4;
#pragma unroll
  for (int s = 0; s < KS; ++s) {
    const int k0 = s * 4 + 2 * hi;
    v2f a, b;
    if (s * 4 + 3 < K) {
      // fully in-range for every lane: straight vector/scalar loads
      a = *(const v2f*)(arow + k0);
      b.x = W[k0 * 16 + nl];
      b.y = W[(k0 + 1) * 16 + nl];
    } else {
      // tail step: clamp addresses, load unconditionally, mask with cndmask
      const int k0c = (k0 < K) ? k0 : 0;
      const int k1c = (k0 + 1 < K) ? (k0 + 1) : 0;
      float a0 = arow[k0c];
      float a1 = arow[k1c];
      float w0 = W[k0c * 16 + nl];
      float w1 = W[k1c * 16 + nl];
      a.x = (k0 < K) ? a0 : 0.0f;
      a.y = (k0 + 1 < K) ? a1 : 0.0f;
      b.x = (k0 < K) ? w0 : 0.0f;
      b.y = (k0 + 1 < K) ? w1 : 0.0f;
    }
    c = __builtin_amdgcn_wmma_f32_16x16x4_f32(
        /*neg_a=*/false, a, /*neg_b=*/false, b,
        /*c_mod=*/(short)0, c, /*reuse_a=*/false, /*reuse_b=*/false);
  }

#pragma unroll
  for (int r = 0; r < 8; ++r) {
    // D layout: M = r + 8*hi, N = nl
    D[((long long)tile * 16 + r + 8 * hi) * 16 + nl] = c[r];
  }
}

// ---------------- edge scatter: out[dst] += in[src] * dinv[src]*dinv[dst] ----
// One thread per (edge, feature); 16 lanes share one 64B feature row, and the
// index loads within a group coalesce to a single request.

__global__ void k_scatter(const long long* __restrict__ src,
                          const long long* __restrict__ dst,
                          const float* __restrict__ dinv,
                          const float* __restrict__ in,
                          float* __restrict__ out, int E) {
  unsigned gid = blockIdx.x * blockDim.x + threadIdx.x;
  int e = (int)(gid >> 4);
  if (e >= E) return;
  int h = (int)(gid & 15u);
  int s = (int)src[e];
  int d = (int)dst[e];
  float nrm = dinv[s] * dinv[d];
  atomicAdd(&out[d * 16 + h], in[s * 16 + h] * nrm);
}

// ---------------- epilogue: self-loop + bias + ReLU ----------------
// out[i,h] = relu(agg[i,h] + xw[i,h]*dinv[i]^2 + bias[h])

__global__ void k_self_bias_relu(const float* __restrict__ agg,
                                 const float* __restrict__ xw,
                                 const float* __restrict__ dinv,
                                 const float* __restrict__ bias,
                                 float* __restrict__ out, int n) {
  unsigned gid = blockIdx.x * blockDim.x + threadIdx.x;
  if (gid >= (unsigned)n * 16u) return;
  int i = (int)(gid >> 4);
  int h = (int)(gid & 15u);
  float di = dinv[i];
  float v = agg[gid] + xw[gid] * di * di + bias[h];
  out[gid] = fmaxf(v, 0.0f);
}

// ---------------- driver ----------------

extern "C" void kernel_launch(void* const* d_in, const int* in_sizes, int n_in,
                              void* d_out, int out_size, void* d_ws, size_t ws_size,
                              hipStream_t stream) {
  const float*     x   = (const float*)d_in[0];
  const long long* ei  = (const long long*)d_in[1];  // int64 [2, E]
  const float*     W1  = (const float*)d_in[2];
  const float*     b1  = (const float*)d_in[3];
  const float*     W2  = (const float*)d_in[4];
  const float*     b2  = (const float*)d_in[5];
  float*           out = (float*)d_out;

  const int N = in_sizes[0] / GCN_F_IN;   // 100000 (multiple of 16)
  const int E = in_sizes[1] / 2;          // 3200000
  const long long* src = ei;
  const long long* dst = ei + E;

  // workspace: [dinv (N, aligned)] [t0 (N*16)] [t1 (N*16)]
  float* ws   = (float*)d_ws;
  float* dinv = ws;
  size_t nAl  = (size_t)((N + 63) & ~63);
  float* t0   = ws + nAl;
  float* t1   = t0 + (size_t)N * 16;

  const int B = 256;
  const int wavesPerBlk = B / 32;
  const int tiles = N / 16;
  const int gemmBlocks = (tiles + wavesPerBlk - 1) / wavesPerBlk;
  const int nhBlocks = (N * 16 + B - 1) / B;
  const unsigned scatBlocks = (unsigned)(((long long)E * 16 + B - 1) / B);

  // degrees -> dinv
  k_init_deg<<<(N + B - 1) / B, B, 0, stream>>>(dinv, N);
  k_count_deg<<<(E + B - 1) / B, B, 0, stream>>>(dst, E, dinv);
  k_rsqrt_inplace<<<(N + B - 1) / B, B, 0, stream>>>(dinv, N);

  // ---- layer 1 ----
  k_gemm_wmma_f32<GCN_F_IN><<<gemmBlocks, B, 0, stream>>>(x, W1, t0, tiles);
  k_zero<<<nhBlocks, B, 0, stream>>>(t1, N * 16);
  k_scatter<<<scatBlocks, B, 0, stream>>>(src, dst, dinv, t0, t1, E);
  k_self_bias_relu<<<nhBlocks, B, 0, stream>>>(t1, t0, dinv, b1, t0, N); // h1 -> t0

  // ---- layer 2 ----
  k_gemm_wmma_f32<GCN_H><<<gemmBlocks, B, 0, stream>>>(t0, W2, t1, tiles); // hw -> t1
  k_zero<<<nhBlocks, B, 0, stream>>>(t0, N * 16);
  k_scatter<<<scatBlocks, B, 0, stream>>>(src, dst, dinv, t1, t0, E);
  k_self_bias_relu<<<nhBlocks, B, 0, stream>>>(t0, t1, dinv, b2, out, N);
}